// SS2D_2911987827302
// MI455X (gfx1250) — compile-verified
//
#include <hip/hip_runtime.h>
#include <hip/hip_bf16.h>
#include <math.h>

// ---------------- problem constants ----------------
#define BB      8
#define HH      128
#define WWID    128
#define DMODEL  96
#define DINNER  96
#define DSTATE  16
#define DTRANK  3
#define NXD     (DTRANK + 2*DSTATE)   // 35
#define LTOT    (HH*WWID)             // 16384
#define TOK     (BB*HH*WWID)          // 131072
#define LNEPS   1e-5f

typedef float v2f __attribute__((ext_vector_type(2)));
typedef float v8f __attribute__((ext_vector_type(8)));

#if defined(__has_builtin)
#  if __has_builtin(__builtin_amdgcn_wmma_f32_16x16x4_f32)
#    define HAVE_WMMA_F32 1
#  endif
#endif
#ifndef HAVE_WMMA_F32
#  define HAVE_WMMA_F32 0
#endif

__device__ __forceinline__ float softplusf(float x) {
    // log(1 + e^x), numerically stable
    if (x > 20.0f) return x;
    return log1pf(__expf(x));
}
__device__ __forceinline__ float siluf(float x) {
    return x * (1.0f / (1.0f + __expf(-x)));
}

// ---------------------------------------------------------------------------
// Kernel 1/6: fp32 WMMA GEMM  C[M,N] = A[M,K] @ W[K,N]   (row-major all)
// K multiple of 4, N multiple of 16, M multiple of 128.
// Each wave computes 16 rows; A fragments (16x4 f32, ISA 7.12.2 layout)
// cached in VGPRs and reused across all N-tiles.
// ---------------------------------------------------------------------------
#if HAVE_WMMA_F32
template <int K, int N>
__global__ void __launch_bounds__(256)
k_gemm_wmma(const float* __restrict__ A, const float* __restrict__ W,
            float* __restrict__ C, int M) {
    const int lane = threadIdx.x & 31;
    const int wave = threadIdx.x >> 5;
    const int r    = lane & 15;     // M (A) / N (B,C) index within tile
    const int half = lane >> 4;     // K-half selector
    const int m0   = (blockIdx.x * 8 + wave) * 16;
    if (m0 >= M) return;            // wave-uniform -> EXEC stays all-ones

    constexpr int KT = K / 4;
    // A fragment cache: lane<16 holds K=kt*4+{0,1}, lane>=16 holds K=kt*4+{2,3}
    v2f afrag[KT];
    const float* arow = A + (size_t)(m0 + r) * K + 2 * half;
#pragma unroll
    for (int kt = 0; kt < KT; ++kt) {
        afrag[kt].x = arow[kt * 4 + 0];
        afrag[kt].y = arow[kt * 4 + 1];
    }

#pragma unroll 1
    for (int n0 = 0; n0 < N; n0 += 16) {
        v8f acc = {};
#pragma unroll
        for (int kt = 0; kt < KT; ++kt) {
            // B fragment 4x16: lane<16 -> K=kt*4+{0,1}; lane>=16 -> K=kt*4+{2,3}
            const float* wp = W + (size_t)(kt * 4 + 2 * half) * N + n0 + r;
            v2f bfrag;
            bfrag.x = wp[0];
            bfrag.y = wp[N];
            acc = __builtin_amdgcn_wmma_f32_16x16x4_f32(
                /*neg_a=*/false, afrag[kt], /*neg_b=*/false, bfrag,
                /*c_mod=*/(short)0, acc, /*reuse_a=*/false, /*reuse_b=*/false);
        }
        // C/D layout: VGPR v, lane<16 -> M=v ; lane>=16 -> M=v+8 ; N=r
        float* cp = C + (size_t)(m0 + 8 * half) * N + n0 + r;
#pragma unroll
        for (int v = 0; v < 8; ++v) cp[(size_t)v * N] = acc[v];
    }
}
#else
// Fallback (compiles everywhere): plain FMA GEMM, one C element per thread.
__global__ void __launch_bounds__(256)
k_gemm_naive(const float* __restrict__ A, const float* __restrict__ W,
             float* __restrict__ C, int M, int N, int K) {
    int idx = blockIdx.x * 256 + threadIdx.x;
    if (idx >= M * N) return;
    int n = idx % N, m = idx / N;
    const float* a = A + (size_t)m * K;
    float s = 0.f;
    for (int k = 0; k < K; ++k) s = fmaf(a[k], W[(size_t)k * N + n], s);
    C[(size_t)m * N + n] = s;
}
#endif

// ---------------------------------------------------------------------------
// Kernel 2/6: depthwise 3x3 conv (cross-correlation, pad=1) + bias + SiLU
// input xc = xz[:, 0:96] (stride 192), output xc_act flat [t*96+c]
// ---------------------------------------------------------------------------
__global__ void __launch_bounds__(256)
k_conv_silu(const float* __restrict__ xz, const float* __restrict__ cw,
            const float* __restrict__ cb, float* __restrict__ out) {
    int idx = blockIdx.x * 256 + threadIdx.x;          // over TOK*DINNER
    if (idx >= TOK * DINNER) return;
    int c = idx % DINNER;
    int t = idx / DINNER;
    int w = t % WWID;
    int h = (t / WWID) % HH;
    int b = t / (WWID * HH);
    float s = cb[c];
#pragma unroll
    for (int kh = 0; kh < 3; ++kh) {
        int ih = h + kh - 1;
        if (ih < 0 || ih >= HH) continue;
#pragma unroll
        for (int kw = 0; kw < 3; ++kw) {
            int iw = w + kw - 1;
            if (iw < 0 || iw >= WWID) continue;
            size_t src = ((size_t)((b * HH + ih) * WWID + iw)) * (2 * DINNER) + c;
            s = fmaf(xz[src], cw[(kh * 3 + kw) * DINNER + c], s);
        }
    }
    out[idx] = siluf(s);
}

// ---------------------------------------------------------------------------
// Kernel 3/6: x_dbl = xc_act @ x_proj_w (35 cols), split dt/B/C,
// delta = softplus(softplus(dt @ dt_w + dt_b))   [reference applies it twice]
// Writes delta flat [t*96+c], Bs/Cs flat [t*16+n].
// ---------------------------------------------------------------------------
__global__ void __launch_bounds__(256)
k_xproj(const float* __restrict__ xc, const float* __restrict__ xpw,
        const float* __restrict__ dtw, const float* __restrict__ dtb,
        float* __restrict__ delta, float* __restrict__ Bsb,
        float* __restrict__ Csb) {
    __shared__ float s_xpw[DINNER * NXD];   // 96*35 floats = 13.1 KB
    __shared__ float s_dtw[DTRANK * DINNER];
    __shared__ float s_dtb[DINNER];
    for (int i = threadIdx.x; i < DINNER * NXD; i += 256) s_xpw[i] = xpw[i];
    for (int i = threadIdx.x; i < DTRANK * DINNER; i += 256) s_dtw[i] = dtw[i];
    for (int i = threadIdx.x; i < DINNER; i += 256) s_dtb[i] = dtb[i];
    __syncthreads();

    int t = blockIdx.x * 256 + threadIdx.x;
    if (t >= TOK) return;
    const float* xv = xc + (size_t)t * DINNER;

    float acc[NXD];
#pragma unroll
    for (int j = 0; j < NXD; ++j) acc[j] = 0.f;
    for (int k = 0; k < DINNER; ++k) {
        float xk = xv[k];
#pragma unroll
        for (int j = 0; j < NXD; ++j) acc[j] = fmaf(xk, s_xpw[k * NXD + j], acc[j]);
    }
    // delta
    for (int c = 0; c < DINNER; ++c) {
        float dv = s_dtb[c];
        dv = fmaf(acc[0], s_dtw[0 * DINNER + c], dv);
        dv = fmaf(acc[1], s_dtw[1 * DINNER + c], dv);
        dv = fmaf(acc[2], s_dtw[2 * DINNER + c], dv);
        dv = softplusf(softplusf(dv));
        delta[(size_t)t * DINNER + c] = dv;
    }
#pragma unroll
    for (int n = 0; n < DSTATE; ++n) {
        Bsb[(size_t)t * DSTATE + n] = acc[DTRANK + n];
        Csb[(size_t)t * DSTATE + n] = acc[DTRANK + DSTATE + n];
    }
}

// ---------------------------------------------------------------------------
// Kernel 4/6: selective scan. 768 independent (b,d) rows; each row = 16 lanes
// (one per state n); 2 rows per wave32. Per-step critical path = one FMA on h.
// y reduced across the 16 state lanes with shfl_xor (off the recurrence path).
// ---------------------------------------------------------------------------
__global__ void __launch_bounds__(256)
k_scan(const float* __restrict__ xc, const float* __restrict__ delta,
       const float* __restrict__ Bsb, const float* __restrict__ Csb,
       const float* __restrict__ A_logs, const float* __restrict__ Ds,
       float* __restrict__ ybuf) {
    const int grp = threadIdx.x >> 4;           // 16 row-groups per block
    const int n   = threadIdx.x & 15;           // state index
    const int row = blockIdx.x * 16 + grp;      // 0..767
    const int b   = row / DINNER;
    const int d   = row % DINNER;

    const float An = -__expf(A_logs[d * DSTATE + n]);   // A = -exp(A_logs)
    const float Dd = Ds[d];

    const float* up = xc    + ((size_t)b * DINNER + d) * LTOT;
    const float* dp = delta + ((size_t)b * DINNER + d) * LTOT;
    const float* bp = Bsb   + ((size_t)b * DSTATE + n) * LTOT;
    const float* cp = Csb   + ((size_t)b * DSTATE + n) * LTOT;
    float*       yp = ybuf  + ((size_t)b * DINNER + d) * LTOT;

    float h = 0.f;
    for (int l = 0; l < LTOT; ++l) {
        if ((l & 63) == 0) {                     // stream-ahead into caches
            __builtin_prefetch(bp + l + 512, 0, 0);
            __builtin_prefetch(cp + l + 512, 0, 0);
            __builtin_prefetch(dp + l + 512, 0, 0);
            __builtin_prefetch(up + l + 512, 0, 0);
        }
        float dl = dp[l];
        float u  = up[l];
        float dA = __expf(dl * An);
        h = fmaf(dA, h, (dl * u) * bp[l]);       // recurrence: single FMA chain
        float py = h * cp[l];
        py += __shfl_xor(py, 1, 16);
        py += __shfl_xor(py, 2, 16);
        py += __shfl_xor(py, 4, 16);
        py += __shfl_xor(py, 8, 16);
        if (n == 0) yp[l] = fmaf(Dd, u, py);
    }
}

// ---------------------------------------------------------------------------
// Kernel 5/6: LayerNorm over d (96, contiguous per token) + SiLU(z) gating
// ---------------------------------------------------------------------------
__global__ void __launch_bounds__(256)
k_ln_gate(const float* __restrict__ ybuf, const float* __restrict__ xz,
          const float* __restrict__ ln_g, const float* __restrict__ ln_b,
          float* __restrict__ g) {
    int t = blockIdx.x * 256 + threadIdx.x;
    if (t >= TOK) return;
    const float* yp = ybuf + (size_t)t * DINNER;
    float s = 0.f, s2 = 0.f;
    for (int c = 0; c < DINNER; ++c) {
        float v = yp[c];
        s += v;
        s2 = fmaf(v, v, s2);
    }
    const float inv_n = 1.0f / (float)DINNER;
    float mu  = s * inv_n;
    float var = fmaf(-mu, mu, s2 * inv_n);
    float rinv = rsqrtf(var + LNEPS);
    const float* zp = xz + (size_t)t * (2 * DINNER) + DINNER;
    float* gp = g + (size_t)t * DINNER;
    for (int c = 0; c < DINNER; ++c) {
        float yn = fmaf((yp[c] - mu) * rinv, ln_g[c], ln_b[c]);
        gp[c] = yn * siluf(zp[c]);
    }
}

// ---------------------------------------------------------------------------
// launch
// ---------------------------------------------------------------------------
extern "C" void kernel_launch(void* const* d_in, const int* in_sizes, int n_in,
                              void* d_out, int out_size, void* d_ws, size_t ws_size,
                              hipStream_t stream) {
    const float* x          = (const float*)d_in[0];
    const float* in_proj_w  = (const float*)d_in[1];
    const float* conv_w     = (const float*)d_in[2];
    const float* conv_b     = (const float*)d_in[3];
    const float* x_proj_w   = (const float*)d_in[4];
    const float* dt_w       = (const float*)d_in[5];
    const float* dt_b       = (const float*)d_in[6];
    const float* A_logs     = (const float*)d_in[7];
    const float* Ds         = (const float*)d_in[8];
    const float* ln_g       = (const float*)d_in[9];
    const float* ln_b       = (const float*)d_in[10];
    const float* out_proj_w = (const float*)d_in[11];
    float* out = (float*)d_out;

    // workspace layout (flat fp32)
    float* ws    = (float*)d_ws;
    float* xz    = ws;                         // TOK*192
    float* xcact = xz    + (size_t)TOK * 192;  // TOK*96
    float* delta = xcact + (size_t)TOK * 96;   // TOK*96
    float* Bsb   = delta + (size_t)TOK * 96;   // TOK*16
    float* Csb   = Bsb   + (size_t)TOK * 16;   // TOK*16
    float* ybuf  = Csb   + (size_t)TOK * 16;   // TOK*96
    float* g     = xcact;                      // alias: xc dead after scan

    // 1) xz = x @ in_proj_w  (131072 x 96 x 192)
#if HAVE_WMMA_F32
    k_gemm_wmma<DMODEL, 2 * DINNER><<<TOK / 128, 256, 0, stream>>>(x, in_proj_w, xz, TOK);
#else
    k_gemm_naive<<<(TOK * 2 * DINNER + 255) / 256, 256, 0, stream>>>(
        x, in_proj_w, xz, TOK, 2 * DINNER, DMODEL);
#endif

    // 2) depthwise conv + SiLU
    k_conv_silu<<<(TOK * DINNER + 255) / 256, 256, 0, stream>>>(xz, conv_w, conv_b, xcact);

    // 3) x_dbl / delta / B / C
    k_xproj<<<TOK / 256, 256, 0, stream>>>(xcact, x_proj_w, dt_w, dt_b, delta, Bsb, Csb);

    // 4) selective scan (768 rows, 16 rows per 256-thread block)
    k_scan<<<(BB * DINNER) / 16, 256, 0, stream>>>(xcact, delta, Bsb, Csb, A_logs, Ds, ybuf);

    // 5) LayerNorm + SiLU gate
    k_ln_gate<<<TOK / 256, 256, 0, stream>>>(ybuf, xz, ln_g, ln_b, g);

    // 6) out = g @ out_proj_w  (131072 x 96 x 96)
#if HAVE_WMMA_F32
    k_gemm_wmma<DINNER, DMODEL><<<TOK / 128, 256, 0, stream>>>(g, out_proj_w, out, TOK);
#else
    k_gemm_naive<<<(TOK * DMODEL + 255) / 256, 256, 0, stream>>>(
        g, out_proj_w, out, TOK, DMODEL, DINNER);
#endif
}